// ModelNet40_12807592477200
// MI455X (gfx1250) — compile-verified
//
#include <hip/hip_runtime.h>
#include <cstdint>
#include <cstddef>

typedef float v2f __attribute__((ext_vector_type(2)));
typedef float v8f __attribute__((ext_vector_type(8)));

#define BATCH 32
#define NPTS  2048
#define NCENT 16

#ifndef __has_builtin
#define __has_builtin(x) 0
#endif

// ---------------------------------------------------------------------------
// Full-precision CDNA5 matrix op: D(16x16,f32) = A(16x4,f32)*B(4x16,f32)+C
//   A: row = lane&15, VGPR0 -> K = 2*(lane>>4), VGPR1 -> K = 2*(lane>>4)+1
//   B: col = lane&15, VGPR0 -> K = 2*(lane>>4), VGPR1 -> K = 2*(lane>>4)+1
//   C: VGPR r -> row = r + 8*(lane>>4), col = lane&15
// ---------------------------------------------------------------------------
__device__ __forceinline__ v8f wmma4(v2f a, v2f b, v8f c) {
  return __builtin_amdgcn_wmma_f32_16x16x4_f32(false, a, false, b, (short)0, c,
                                               false, false);
}

// ---------------------------------------------------------------------------
// Tensor Data Mover: stage a 2-D tile (tile_w dwords x tile_h rows, row
// stride stride_dw dwords) from global into LDS.  D# built per ISA ch.8.
// ---------------------------------------------------------------------------
#if defined(__HIP_DEVICE_COMPILE__) &&                                        \
    __has_builtin(__builtin_amdgcn_tensor_load_to_lds) &&                     \
    __has_builtin(__builtin_amdgcn_s_wait_tensorcnt)
#define HAVE_TDM 1
#else
#define HAVE_TDM 0
#endif

#if HAVE_TDM
typedef unsigned int u32x4 __attribute__((ext_vector_type(4)));
typedef int i32x4 __attribute__((ext_vector_type(4)));
typedef int i32x8 __attribute__((ext_vector_type(8)));

__device__ __forceinline__ void tdm_load_2d(const float* gsrc, float* lds,
                                            unsigned tile_w_dw, unsigned tile_h,
                                            unsigned stride_dw) {
  unsigned long long ga = (unsigned long long)(const void*)gsrc;
  unsigned lo = (unsigned)(unsigned long long)(void*)lds;  // LDS = addr[31:0]
  // ---- D# group 0 (128b): count=1 | lds_addr | global_addr[56:0] | type=2
  u32x4 g0;
  g0[0] = 1u;
  g0[1] = lo;
  g0[2] = (unsigned)(ga & 0xffffffffu);
  g0[3] = (unsigned)((ga >> 32) & 0x01ffffffu) | (2u << 30);
  // ---- D# group 1 (256b)
  unsigned dim0 = stride_dw;        // tensor extent dim0 (>= tile_w)
  unsigned dim1 = 0x00ffffffu;      // generous dim1 extent (tiles always valid)
  i32x8 g1;
  g1[0] = (int)(2u << 16);                                   // data_size = 4B
  g1[1] = (int)((dim0 & 0xffffu) << 16);                     // dim0[15:0]
  g1[2] = (int)(((dim0 >> 16) & 0xffffu) |                   // dim0[31:16]
                ((dim1 & 0xffffu) << 16));                   // dim1[15:0]
  g1[3] = (int)(((dim1 >> 16) & 0xffffu) |                   // dim1[31:16]
                ((tile_w_dw & 0xffffu) << 16));              // tile_dim0
  g1[4] = (int)(tile_h & 0xffffu);                           // tile_dim1
  g1[5] = (int)stride_dw;                                    // dim0_stride lo32
  g1[6] = 0;                                                 // hi16 + dim1_stride
  g1[7] = 0;
  i32x4 z = {0, 0, 0, 0};
#if __has_include(<hip/amd_detail/amd_gfx1250_TDM.h>)
  i32x8 z8 = {0, 0, 0, 0, 0, 0, 0, 0};
  __builtin_amdgcn_tensor_load_to_lds(g0, g1, z, z, z8, 0);  // clang-23 form
#else
  __builtin_amdgcn_tensor_load_to_lds(g0, g1, z, z, 0);      // ROCm 7.2 form
#endif
}
#endif  // HAVE_TDM

// ---------------------------------------------------------------------------
// KNN: one thread per query, register-resident unrolled insertion top-K.
// ---------------------------------------------------------------------------
template <int K>
__global__ void knn_kernel(const float* __restrict__ pts, int M, int Nin,
                           int* __restrict__ idx_out) {
  int t = blockIdx.x * blockDim.x + threadIdx.x;
  if (t >= BATCH * M) return;
  int b = t / M, m = t % M;
  const float* P = pts + (size_t)b * NPTS * 3;
  float qx = P[m * 3 + 0], qy = P[m * 3 + 1], qz = P[m * 3 + 2];

  float bd[K];
  int bi[K];
#pragma unroll
  for (int j = 0; j < K; ++j) { bd[j] = 3.4e38f; bi[j] = 0; }

  for (int n = 0; n < Nin; ++n) {
    float dx = P[n * 3 + 0] - qx;
    float dy = P[n * 3 + 1] - qy;
    float dz = P[n * 3 + 2] - qz;
    float d = dx * dx + dy * dy + dz * dz;
    if (d < bd[K - 1]) {
      bd[K - 1] = d;
      bi[K - 1] = n;
#pragma unroll
      for (int j = K - 1; j > 0; --j) {
        if (bd[j] < bd[j - 1]) {
          float td = bd[j]; bd[j] = bd[j - 1]; bd[j - 1] = td;
          int ti = bi[j];  bi[j] = bi[j - 1];  bi[j - 1] = ti;
        }
      }
    }
  }
  int* o = idx_out + (size_t)t * K;
#pragma unroll
  for (int j = 0; j < K; ++j) o[j] = bi[j];
}

// ---------------------------------------------------------------------------
// Per-(b,m) neighborhood kernel: gather -> normalize -> RBF feature (K x 48)
// -> MLP 48->32->16->16 (fp32 WMMA) -> einsum nfeat^T(CINxK) * h(Kx16)
// One wave32 per block; weight matrices staged via TDM.
// ---------------------------------------------------------------------------
template <int K, int CIN>
__global__ void __launch_bounds__(32)
point_mlp_einsum(const float* __restrict__ pts,      // (B, NPTS, 3)
                 const float* __restrict__ feat_in,  // (B, Nin, CIN)
                 const int* __restrict__ knn,        // (B*M, K)
                 const float* __restrict__ centers,  // (3, 16)
                 const float* __restrict__ l1w, const float* __restrict__ l1b,
                 const float* __restrict__ l2w, const float* __restrict__ l2b,
                 const float* __restrict__ l3w, const float* __restrict__ l3b,
                 int M, int Nin,
                 float* __restrict__ fout)           // (B*M, CIN*16)
{
  constexpr int CINP = ((CIN + 15) / 16) * 16;
  __shared__ float s_d [K * 48];
  __shared__ float s_h1[K * 32];
  __shared__ float s_h2[K * 16];
  __shared__ float s_h [K * 16];
  __shared__ float s_w1[48 * 32];
  __shared__ float s_w2[32 * 16];
  __shared__ float s_w3[16 * 16];
  __shared__ float s_nf[K * CINP];
  __shared__ int   s_idx[K];

  const int bm = blockIdx.x;
  const int b = bm / M, m = bm % M;
  const int lane = threadIdx.x;
  const int half = lane >> 4, col = lane & 15;

  if (lane < K) s_idx[lane] = knn[(size_t)bm * K + lane];
#if HAVE_TDM
  tdm_load_2d(l1w, s_w1, 48 * 32, 1, 48 * 32);   // contiguous weight rows
  tdm_load_2d(l2w, s_w2, 32 * 16, 1, 32 * 16);
  tdm_load_2d(l3w, s_w3, 16 * 16, 1, 16 * 16);
  __builtin_amdgcn_s_wait_tensorcnt(0);
#else
  for (int t = lane; t < 48 * 32; t += 32) s_w1[t] = l1w[t];
  for (int t = lane; t < 32 * 16; t += 32) s_w2[t] = l2w[t];
  for (int t = lane; t < 16 * 16; t += 32) s_w3[t] = l3w[t];
#endif
  __syncthreads();

  // ---- gather neighbor pts, max-norm normalize, build 48-dim RBF feature
  const float* P = pts + (size_t)b * NPTS * 3;
  float nx = 0.f, ny = 0.f, nz = 0.f, nrm2 = 0.f;
  if (lane < K) {
    int r = s_idx[lane];
    nx = P[r * 3 + 0] - P[m * 3 + 0];
    ny = P[r * 3 + 1] - P[m * 3 + 1];
    nz = P[r * 3 + 2] - P[m * 3 + 2];
    nrm2 = nx * nx + ny * ny + nz * nz;
  }
#pragma unroll
  for (int off = 16; off; off >>= 1)
    nrm2 = fmaxf(nrm2, __shfl_xor(nrm2, off, 32));
  float maxi = sqrtf(nrm2);
  maxi = (maxi == 0.f) ? 1.f : maxi;
  float inv = 1.f / maxi;
  nx *= inv; ny *= inv; nz *= inv;
  if (lane < K) {
#pragma unroll
    for (int j = 0; j < NCENT; ++j) {
      s_d[lane * 48 +      j] = nx - centers[     j];
      s_d[lane * 48 + 16 + j] = ny - centers[16 + j];
      s_d[lane * 48 + 32 + j] = nz - centers[32 + j];
    }
  }
  __syncthreads();

  // ---- MLP1: (K x 48) @ (48 x 32) + b, ReLU -> s_h1
  for (int rt = 0; rt < K / 16; ++rt)
    for (int ct = 0; ct < 2; ++ct) {
      v8f c;
      float bv = l1b[ct * 16 + col];
#pragma unroll
      for (int r = 0; r < 8; ++r) c[r] = bv;
      for (int k0 = 0; k0 < 48; k0 += 4) {
        v2f a, bq;
        a.x  = s_d[(rt * 16 + col) * 48 + k0 + 2 * half];
        a.y  = s_d[(rt * 16 + col) * 48 + k0 + 2 * half + 1];
        bq.x = s_w1[(k0 + 2 * half) * 32 + ct * 16 + col];
        bq.y = s_w1[(k0 + 2 * half + 1) * 32 + ct * 16 + col];
        c = wmma4(a, bq, c);
      }
#pragma unroll
      for (int r = 0; r < 8; ++r)
        s_h1[(rt * 16 + r + 8 * half) * 32 + ct * 16 + col] = fmaxf(c[r], 0.f);
    }
  __syncthreads();

  // ---- MLP2: (K x 32) @ (32 x 16) + b, ReLU -> s_h2
  for (int rt = 0; rt < K / 16; ++rt) {
    v8f c;
    float bv = l2b[col];
#pragma unroll
    for (int r = 0; r < 8; ++r) c[r] = bv;
    for (int k0 = 0; k0 < 32; k0 += 4) {
      v2f a, bq;
      a.x  = s_h1[(rt * 16 + col) * 32 + k0 + 2 * half];
      a.y  = s_h1[(rt * 16 + col) * 32 + k0 + 2 * half + 1];
      bq.x = s_w2[(k0 + 2 * half) * 16 + col];
      bq.y = s_w2[(k0 + 2 * half + 1) * 16 + col];
      c = wmma4(a, bq, c);
    }
#pragma unroll
    for (int r = 0; r < 8; ++r)
      s_h2[(rt * 16 + r + 8 * half) * 16 + col] = fmaxf(c[r], 0.f);
  }
  __syncthreads();

  // ---- MLP3: (K x 16) @ (16 x 16) + b, ReLU -> s_h
  for (int rt = 0; rt < K / 16; ++rt) {
    v8f c;
    float bv = l3b[col];
#pragma unroll
    for (int r = 0; r < 8; ++r) c[r] = bv;
    for (int k0 = 0; k0 < 16; k0 += 4) {
      v2f a, bq;
      a.x  = s_h2[(rt * 16 + col) * 16 + k0 + 2 * half];
      a.y  = s_h2[(rt * 16 + col) * 16 + k0 + 2 * half + 1];
      bq.x = s_w3[(k0 + 2 * half) * 16 + col];
      bq.y = s_w3[(k0 + 2 * half + 1) * 16 + col];
      c = wmma4(a, bq, c);
    }
#pragma unroll
    for (int r = 0; r < 8; ++r)
      s_h[(rt * 16 + r + 8 * half) * 16 + col] = fmaxf(c[r], 0.f);
  }
  __syncthreads();

  // ---- gather nfeat (K x CIN) into LDS (zero-padded to CINP)
  for (int t = lane; t < K * CINP; t += 32) s_nf[t] = 0.f;
  __syncthreads();
  const float* FB = feat_in + (size_t)b * Nin * CIN;
  for (int k = 0; k < K; ++k) {
    const float* F = FB + (size_t)s_idx[k] * CIN;
    for (int c2 = lane; c2 < CIN; c2 += 32) s_nf[k * CINP + c2] = F[c2];
  }
  __syncthreads();

  // ---- einsum: f(CIN x 16) = nfeat^T (CIN x K) * h (K x 16)
  for (int ct = 0; ct < CINP / 16; ++ct) {
    v8f c = {};
    for (int k0 = 0; k0 < K; k0 += 4) {
      v2f a, bq;
      a.x  = s_nf[(k0 + 2 * half) * CINP + ct * 16 + col];  // transposed A
      a.y  = s_nf[(k0 + 2 * half + 1) * CINP + ct * 16 + col];
      bq.x = s_h[(k0 + 2 * half) * 16 + col];
      bq.y = s_h[(k0 + 2 * half + 1) * 16 + col];
      c = wmma4(a, bq, c);
    }
#pragma unroll
    for (int r = 0; r < 8; ++r) {
      int crow = ct * 16 + r + 8 * half;
      if (crow < CIN)
        fout[(size_t)bm * CIN * 16 + (size_t)crow * 16 + col] = c[r];
    }
  }
}

// ---------------------------------------------------------------------------
// GEMM + (scale, bias, BN, ReLU). 4 waves/block; each wave owns one 16-col
// panel; the 16x16 A-chunk is staged in LDS via TDM (wave 0) per k-step.
// Column loads are CLAMPED (dead columns never stored) so the inner loop has
// no divergence and EXEC stays all-ones for WMMA.
// ---------------------------------------------------------------------------
__global__ void __launch_bounds__(128)
gemm_bias_bn_relu(const float* __restrict__ A,   // (rows, J)
                  const float* __restrict__ W,   // (J, C)
                  const float* __restrict__ bias,
                  const float* __restrict__ bn_g, const float* __restrict__ bn_b,
                  const float* __restrict__ bn_m, const float* __restrict__ bn_v,
                  float* __restrict__ out, int rows, int J, int C,
                  float scale, int relu) {
  (void)rows;
  __shared__ float sA[16 * 16];
  const int tid = threadIdx.x;
  const int lane = tid & 31, wave = tid >> 5;
  const int half = lane >> 4, col = lane & 15;
  const int row0 = blockIdx.x * 16;
  const int ct = blockIdx.y * 4 + wave;
  const int cc = ct * 16 + col;
  const int ccl = (cc < C) ? cc : (C - 1);  // clamp: garbage cols never stored

  v8f c = {};
  for (int kc = 0; kc < J; kc += 16) {
    __syncthreads();  // previous chunk fully consumed
#if HAVE_TDM
    if (wave == 0) {
      tdm_load_2d(A + (size_t)row0 * J + kc, sA, 16, 16, (unsigned)J);
      __builtin_amdgcn_s_wait_tensorcnt(0);
    }
#else
    {
      int e = tid * 2, r = e >> 4, k = e & 15;
      sA[e]     = A[(size_t)(row0 + r) * J + kc + k];
      sA[e + 1] = A[(size_t)(row0 + r) * J + kc + k + 1];
    }
#endif
    __syncthreads();
    __builtin_prefetch(W + (size_t)(kc + 16) * C + ct * 16, 0, 1);
#pragma unroll
    for (int k0 = 0; k0 < 16; k0 += 4) {
      v2f a, bq;
      a.x = sA[col * 16 + k0 + 2 * half];
      a.y = sA[col * 16 + k0 + 2 * half + 1];
      int kr = kc + k0 + 2 * half;
      bq.x = W[(size_t)kr * C + ccl];
      bq.y = W[(size_t)(kr + 1) * C + ccl];
      c = wmma4(a, bq, c);
    }
  }

  if (cc < C) {
    float bv = bias[cc];
    float g = 1.f, bb = 0.f, mm = 0.f, iv = 1.f;
    if (bn_g) {
      mm = bn_m[cc];
      iv = rsqrtf(bn_v[cc] + 1e-5f);
      g = bn_g[cc];
      bb = bn_b[cc];
    }
#pragma unroll
    for (int r = 0; r < 8; ++r) {
      int rr = row0 + r + 8 * half;
      float v = c[r] * scale + bv;
      if (bn_g) v = (v - mm) * iv * g + bb;
      if (relu) v = fmaxf(v, 0.f);
      out[(size_t)rr * C + cc] = v;
    }
  }
}

// ---------------------------------------------------------------------------
// Host launch
// ---------------------------------------------------------------------------
extern "C" void kernel_launch(void* const* d_in, const int* in_sizes, int n_in,
                              void* d_out, int out_size, void* d_ws,
                              size_t ws_size, hipStream_t stream) {
  (void)in_sizes; (void)n_in; (void)out_size; (void)ws_size;

  // setup_inputs dict order: x, input_pts, params (pytree sorted keys:
  // fc_b, fc_w, layers[0..4] each: b, bn_b, bn_g, bn_m, bn_v, centers,
  // l1b, l1w, l2b, l2w, l3b, l3w, w)
  const float* x    = (const float*)d_in[0];
  const float* pts  = (const float*)d_in[1];
  const float* fc_b = (const float*)d_in[2];
  const float* fc_w = (const float*)d_in[3];
  const float* LP[5][13];
  for (int l = 0; l < 5; ++l)
    for (int k = 0; k < 13; ++k) LP[l][k] = (const float*)d_in[4 + l * 13 + k];
  enum { P_B = 0, P_BNB, P_BNG, P_BNM, P_BNV, P_CTR, P_L1B, P_L1W,
         P_L2B, P_L2W, P_L3B, P_L3W, P_W };

  // workspace layout (~50 MB)
  char* ws = (char*)d_ws;
  int*   idxbuf = (int*)ws;                                   // 32*1024*32 ints
  float* fbuf   = (float*)(ws + (size_t)32 * 1024 * 32 * 4);  // up to 8.4M floats
  float* featA  = fbuf + (size_t)8388608;                     // up to 2.1M floats
  float* featB  = featA + (size_t)2097152;                    // up to 1.05M floats

  struct { int Cin, Cout, K, M, Nin; } cfg[5] = {
      {1, 64, 32, 1024, 2048}, {64, 128, 32, 256, 1024},
      {128, 256, 16, 64, 256}, {256, 256, 16, 16, 64},
      {256, 512, 16, 1, 16}};

  const float* fin = x;
  float* feats[2] = {featA, featB};

  for (int l = 0; l < 5; ++l) {
    const int Cin = cfg[l].Cin, Cout = cfg[l].Cout;
    const int K = cfg[l].K, M = cfg[l].M, Nin = cfg[l].Nin;
    const int total = BATCH * M;

    // 1) KNN
    if (K == 32)
      knn_kernel<32><<<(total + 255) / 256, 256, 0, stream>>>(pts, M, Nin, idxbuf);
    else
      knn_kernel<16><<<(total + 255) / 256, 256, 0, stream>>>(pts, M, Nin, idxbuf);

    // 2) gather + weight-MLP + einsum (one wave per (b,m))
#define MLP_ARGS pts, fin, idxbuf, LP[l][P_CTR], LP[l][P_L1W], LP[l][P_L1B],  \
                 LP[l][P_L2W], LP[l][P_L2B], LP[l][P_L3W], LP[l][P_L3B],      \
                 M, Nin, fbuf
    switch (l) {
      case 0: point_mlp_einsum<32, 1  ><<<total, 32, 0, stream>>>(MLP_ARGS); break;
      case 1: point_mlp_einsum<32, 64 ><<<total, 32, 0, stream>>>(MLP_ARGS); break;
      case 2: point_mlp_einsum<16, 128><<<total, 32, 0, stream>>>(MLP_ARGS); break;
      case 3: point_mlp_einsum<16, 256><<<total, 32, 0, stream>>>(MLP_ARGS); break;
      case 4: point_mlp_einsum<16, 256><<<total, 32, 0, stream>>>(MLP_ARGS); break;
    }
#undef MLP_ARGS

    // 3) out = f @ w / K + b ; BN ; ReLU
    float* fo = feats[l & 1];
    int ctiles = (Cout + 15) / 16;
    dim3 grid(total / 16, (ctiles + 3) / 4);
    gemm_bias_bn_relu<<<grid, 128, 0, stream>>>(
        fbuf, LP[l][P_W], LP[l][P_B], LP[l][P_BNG], LP[l][P_BNB],
        LP[l][P_BNM], LP[l][P_BNV], fo, total, Cin * 16, Cout,
        1.0f / (float)K, 1);
    fin = fo;
  }

  // final FC: (32, 512) @ (512, 40) + fc_b
  dim3 g(32 / 16, 1);  // 3 column tiles fit in one 4-wave block
  gemm_bias_bn_relu<<<g, 128, 0, stream>>>(fin, fc_w, fc_b, nullptr, nullptr,
                                           nullptr, nullptr, (float*)d_out, 32,
                                           512, 40, 1.0f, 0);
}